// SurfaceCodeGNN_76596446756908
// MI455X (gfx1250) — compile-verified
//
#include <hip/hip_runtime.h>
#include <hip/hip_bf16.h>
#include <math.h>

#define HD 128
#define BN_EPS 1e-5f

typedef __attribute__((ext_vector_type(2))) float v2f;
typedef __attribute__((ext_vector_type(8))) float v8f;

// Fast sigmoid/SiLU: v_exp_f32 + v_rcp_f32 (no precise-divide fixup chains).
__device__ __forceinline__ float fast_sigmoid(float y) {
    return __builtin_amdgcn_rcpf(1.0f + __expf(-y));
}
__device__ __forceinline__ float fast_silu(float y) {
    return y * fast_sigmoid(y);
}

// ---------------------------------------------------------------------------
// Degree / normalization
// ---------------------------------------------------------------------------
__global__ void k_init_deg(float* __restrict__ deg, int N) {
    int i = blockIdx.x * blockDim.x + threadIdx.x;
    if (i < N) deg[i] = 1.0f;  // self-loop contributes 1 to every node
}

__global__ void k_accum_deg(const long long* __restrict__ ei, float* __restrict__ deg, int E) {
    int e = blockIdx.x * blockDim.x + threadIdx.x;
    if (e < E) {
        long long c = ei[(long long)E + e];  // col = target index
        atomicAdd(&deg[c], 1.0f);
    }
}

__global__ void k_deg_to_dis(float* __restrict__ deg, int N) {
    int i = blockIdx.x * blockDim.x + threadIdx.x;
    if (i < N) deg[i] = rsqrtf(deg[i]);  // deg >= 1 always (self loops)
}

// ---------------------------------------------------------------------------
// Layer 0: outer product x[N,1] * W0[1,128], fused with self-loop agg init:
//   hW[i,j]  = x[i] * W0[j]
//   agg[i,j] = dis[i]^2 * hW[i,j] + bias[j]
// ---------------------------------------------------------------------------
__global__ void k_layer0_fused(const float* __restrict__ x, const float* __restrict__ W0,
                               const float* __restrict__ dis, const float* __restrict__ bias,
                               float* __restrict__ hW, float* __restrict__ agg, int total) {
    int idx = blockIdx.x * blockDim.x + threadIdx.x;
    if (idx < total) {
        int i = idx >> 7;
        int j = idx & (HD - 1);
        float v = x[i] * W0[j];
        float d = dis[i];
        hW[idx] = v;
        agg[idx] = d * d * v + bias[j];
    }
}

// ---------------------------------------------------------------------------
// Fused BN+SiLU + WMMA fp32 GEMM + self-loop epilogue.
//   a[i,k]     = silu(scl[k]*Araw[i,k] + shf[k])        (BN of previous layer)
//   C[N,128]   = a @ W[128,128]
//   agg[N,128] = dis^2 * C + bias                        (self-loop init)
// block = 256 threads = 8 waves; wave w owns rows [m0,m0+16) with
// m0 = 128*blockIdx + 16*w, and ALL 8 column tiles (v8f acc[8]).
// A fragment loaded once per k-step -> 8 back-to-back V_WMMA_F32_16X16X4_F32.
// Safe in-place: wave reads only its own 16 rows of A(=agg buffer), fully
// consumed into acc before the epilogue rewrites those same rows.
// ---------------------------------------------------------------------------
__global__ void k_gemm_wmma_bn(const float* __restrict__ A, const float* __restrict__ W,
                               const float* __restrict__ scl, const float* __restrict__ shf,
                               const float* __restrict__ dis, const float* __restrict__ bias,
                               float* __restrict__ C, float* __restrict__ agg, int N) {
    const int wave = threadIdx.x >> 5;
    const int lane = threadIdx.x & 31;
    const int hi = lane >> 4;       // 0: K pair {k,k+1}, 1: {k+2,k+3}
    const int lo = lane & 15;
    const int m0 = blockIdx.x * 128 + wave * 16;

    int arow = m0 + lo;
    if (arow > N - 1) arow = N - 1;  // clamp loads; EXEC stays all-ones for WMMA

    v8f acc[8] = {};
    const float* ap = A + (size_t)arow * HD;

    for (int k = 0; k < HD; k += 4) {
        const int kk = k + 2 * hi;
        // load raw A pair, apply BN (scale/shift) + SiLU on the fly
        v2f raw = *(const v2f*)(ap + kk);
        v2f sc  = *(const v2f*)(scl + kk);
        v2f sh  = *(const v2f*)(shf + kk);
        v2f a;
        a.x = fast_silu(sc.x * raw.x + sh.x);
        a.y = fast_silu(sc.y * raw.y + sh.y);

        const float* bp = W + (size_t)kk * HD + lo;
        #pragma unroll
        for (int t = 0; t < 8; ++t) {
            v2f b;
            b.x = bp[t * 16];
            b.y = bp[t * 16 + HD];
            acc[t] = __builtin_amdgcn_wmma_f32_16x16x4_f32(
                /*neg_a=*/false, a, /*neg_b=*/false, b,
                /*c_mod=*/(short)0, acc[t], /*reuse_a=*/false, /*reuse_b=*/false);
        }
    }

    if (m0 + 16 <= N) {
        // fast path: straight-line stores, no divergence
        #pragma unroll
        for (int r = 0; r < 8; ++r) {
            int row = m0 + r + 8 * hi;   // C/D layout: VGPR r -> M = r (+8 for hi lanes)
            float d2 = dis[row]; d2 *= d2;
            size_t rowo = (size_t)row * HD + lo;
            #pragma unroll
            for (int t = 0; t < 8; ++t) {
                size_t o = rowo + t * 16;
                C[o] = acc[t][r];
                agg[o] = d2 * acc[t][r] + bias[t * 16 + lo];
            }
        }
    } else if (m0 < N) {
        #pragma unroll
        for (int r = 0; r < 8; ++r) {
            int row = m0 + r + 8 * hi;
            if (row < N) {
                float d2 = dis[row]; d2 *= d2;
                size_t rowo = (size_t)row * HD + lo;
                #pragma unroll
                for (int t = 0; t < 8; ++t) {
                    size_t o = rowo + t * 16;
                    C[o] = acc[t][r];
                    agg[o] = d2 * acc[t][r] + bias[t * 16 + lo];
                }
            }
        }
    }
}

// ---------------------------------------------------------------------------
// Edge scatter: one wave per edge; 32 lanes x float4 = 128 features.
// agg[col,:] += dis[row]*dis[col] * hW[row,:]
// ---------------------------------------------------------------------------
__global__ void k_scatter(const float* __restrict__ hW, const float* __restrict__ dis,
                          const long long* __restrict__ ei, float* __restrict__ agg, int E) {
    int e = blockIdx.x * 8 + (threadIdx.x >> 5);
    if (e >= E) return;
    int lane = threadIdx.x & 31;
    long long r = ei[e];
    long long c = ei[(long long)E + e];
    float nrm = dis[r] * dis[c];
    const float4 v = *(const float4*)(hW + (size_t)r * HD + lane * 4);
    float* dst = agg + (size_t)c * HD + lane * 4;
    atomicAdd(dst + 0, nrm * v.x);
    atomicAdd(dst + 1, nrm * v.y);
    atomicAdd(dst + 2, nrm * v.z);
    atomicAdd(dst + 3, nrm * v.w);
}

// ---------------------------------------------------------------------------
// BatchNorm statistics: per-column sum / sumsq, then fold into scale/shift
// ---------------------------------------------------------------------------
__global__ void k_zero(float* __restrict__ p, int n) {
    int i = blockIdx.x * blockDim.x + threadIdx.x;
    if (i < n) p[i] = 0.0f;
}

__global__ void k_bn_stats(const float* __restrict__ h, float* __restrict__ sum,
                           float* __restrict__ sumsq, int N) {
    int c = threadIdx.x;  // 128 threads
    float s = 0.0f, ss = 0.0f;
    for (int i = blockIdx.x; i < N; i += gridDim.x) {
        float v = h[(size_t)i * HD + c];
        s += v;
        ss += v * v;
    }
    atomicAdd(&sum[c], s);
    atomicAdd(&sumsq[c], ss);
}

// scl[c] = gamma*rsqrt(var+eps) ; shf[c] = beta - mu*scl   (so y = scl*h + shf)
__global__ void k_prep_bn(const float* __restrict__ sum, const float* __restrict__ sumsq,
                          const float* __restrict__ gamma, const float* __restrict__ beta,
                          float invN, float* __restrict__ scl, float* __restrict__ shf) {
    int c = threadIdx.x;  // HD threads
    float mu = sum[c] * invN;
    float var = sumsq[c] * invN - mu * mu;
    float s = gamma[c] * rsqrtf(var + BN_EPS);
    scl[c] = s;
    shf[c] = beta[c] - mu * s;
}

// Last layer: BN + SiLU fused directly into segment pooling (skip writing h).
__global__ void k_bn_apply_pool(const float* __restrict__ h, const float* __restrict__ sum,
                                const float* __restrict__ sumsq, const float* __restrict__ gamma,
                                const float* __restrict__ beta, float invN,
                                const long long* __restrict__ batch,
                                float* __restrict__ psums, float* __restrict__ pcnts, int total) {
    int idx = blockIdx.x * blockDim.x + threadIdx.x;
    if (idx < total) {
        int i = idx >> 7;
        int c = idx & (HD - 1);
        float mu = sum[c] * invN;
        float var = sumsq[c] * invN - mu * mu;
        float hn = (h[idx] - mu) * rsqrtf(var + BN_EPS);
        float y = fast_silu(hn * gamma[c] + beta[c]);
        int g = (int)batch[i];
        atomicAdd(&psums[(size_t)g * HD + c], y);
        if (c == 0) atomicAdd(&pcnts[g], 1.0f);
    }
}

// ---------------------------------------------------------------------------
// Head: pooled @ fc1 (+SiLU) @ fc2 (+sigmoid). One 64-thread block per graph.
// ---------------------------------------------------------------------------
__global__ void k_head(const float* __restrict__ sums, const float* __restrict__ cnts,
                       const float* __restrict__ fc1_w, const float* __restrict__ fc1_b,
                       const float* __restrict__ fc2_w, const float* __restrict__ fc2_b,
                       float* __restrict__ out) {
    int g = blockIdx.x;
    int j = threadIdx.x;  // 64
    float inv = __builtin_amdgcn_rcpf(fmaxf(cnts[g], 1.0f));
    float acc = fc1_b[j];
    #pragma unroll 8
    for (int k = 0; k < HD; ++k)
        acc += (sums[(size_t)g * HD + k] * inv) * fc1_w[k * 64 + j];
    float z = fast_silu(acc);

    __shared__ float red[64];
    red[j] = z * fc2_w[j];
    __syncthreads();
    #pragma unroll
    for (int s = 32; s > 0; s >>= 1) {
        if (j < s) red[j] += red[j + s];
        __syncthreads();
    }
    if (j == 0) {
        float o = red[0] + fc2_b[0];
        out[g] = fast_sigmoid(o);
    }
}

// ---------------------------------------------------------------------------
// Host orchestration
// ---------------------------------------------------------------------------
extern "C" void kernel_launch(void* const* d_in, const int* in_sizes, int n_in,
                              void* d_out, int out_size, void* d_ws, size_t ws_size,
                              hipStream_t stream) {
    const float*     x      = (const float*)d_in[0];
    const long long* ei     = (const long long*)d_in[1];
    const long long* batch  = (const long long*)d_in[2];
    const float*     W0     = (const float*)d_in[4];
    const float*     Ws     = (const float*)d_in[5];
    const float*     biases = (const float*)d_in[6];
    const float*     gammas = (const float*)d_in[7];
    const float*     betas  = (const float*)d_in[8];
    const float*     fc1_w  = (const float*)d_in[9];
    const float*     fc1_b  = (const float*)d_in[10];
    const float*     fc2_w  = (const float*)d_in[11];
    const float*     fc2_b  = (const float*)d_in[12];
    float*           out    = (float*)d_out;

    const int N = in_sizes[0];          // 100000
    const int E = in_sizes[1] / 2;      // 600000
    const int G = out_size;             // 256
    const int total = N * HD;

    // Workspace layout (256B-aligned offsets)
    char* ws = (char*)d_ws;
    size_t off = 0;
    auto carve = [&](size_t bytes) { char* p = ws + off; off = (off + bytes + 255) & ~(size_t)255; return p; };
    float* hbuf   = (float*)carve((size_t)total * sizeof(float));   // agg / activations
    float* tbuf   = (float*)carve((size_t)total * sizeof(float));   // h @ W
    float* dis    = (float*)carve((size_t)N * sizeof(float));
    float* stats  = (float*)carve(2 * HD * sizeof(float));          // colsum | colsumsq
    float* scl    = (float*)carve(HD * sizeof(float));              // BN folded scale
    float* shf    = (float*)carve(HD * sizeof(float));              // BN folded shift
    float* psums  = (float*)carve((size_t)G * HD * sizeof(float));
    float* pcnts  = (float*)carve((size_t)G * sizeof(float));
    (void)ws_size; (void)n_in;

    const int TPB = 256;
    const float invN = 1.0f / (float)N;

    // degrees -> dis = deg^-1/2
    k_init_deg<<<(N + TPB - 1) / TPB, TPB, 0, stream>>>(dis, N);
    k_accum_deg<<<(E + TPB - 1) / TPB, TPB, 0, stream>>>(ei, dis, E);
    k_deg_to_dis<<<(N + TPB - 1) / TPB, TPB, 0, stream>>>(dis, N);

    const int elemBlocks = (total + TPB - 1) / TPB;
    const int gemmBlocks = (N + 127) / 128;   // 8 waves/block, 16 rows/wave

    // zero pooled accumulators up front (psums, pcnts contiguous)
    k_zero<<<(G * HD + G + TPB - 1) / TPB, TPB, 0, stream>>>(psums, G * HD + G);

    for (int l = 0; l < 4; ++l) {
        // GEMM (or outer product) with fused input-BN+SiLU and fused
        // self-loop + bias epilogue: tbuf = silu(bn(h)) @ W ; hbuf = dis^2*tbuf + bias
        if (l == 0) {
            k_layer0_fused<<<elemBlocks, TPB, 0, stream>>>(x, W0, dis, biases, tbuf, hbuf, total);
        } else {
            k_gemm_wmma_bn<<<gemmBlocks, 256, 0, stream>>>(hbuf, Ws + (size_t)(l - 1) * HD * HD,
                                                           scl, shf, dis, biases + l * HD,
                                                           tbuf, hbuf, N);
        }

        // edge scatter-add into hbuf
        k_scatter<<<(E + 7) / 8, 256, 0, stream>>>(tbuf, dis, ei, hbuf, E);

        // batch-norm statistics over aggregated features
        k_zero<<<1, 2 * HD, 0, stream>>>(stats, 2 * HD);
        k_bn_stats<<<512, HD, 0, stream>>>(hbuf, stats, stats + HD, N);

        if (l < 3) {
            // fold BN params; consumed by the NEXT layer's fused GEMM
            k_prep_bn<<<1, HD, 0, stream>>>(stats, stats + HD,
                                            gammas + l * HD, betas + l * HD, invN, scl, shf);
        } else {
            // last layer: fuse BN+SiLU with segment-mean pooling accumulation
            k_bn_apply_pool<<<elemBlocks, TPB, 0, stream>>>(hbuf, stats, stats + HD,
                                                            gammas + l * HD, betas + l * HD, invN,
                                                            batch, psums, pcnts, total);
        }
    }

    k_head<<<G, 64, 0, stream>>>(psums, pcnts, fc1_w, fc1_b, fc2_w, fc2_b, out);
}